// GRACE_76398878261380
// MI455X (gfx1250) — compile-verified
//
#include <hip/hip_runtime.h>
#include <hip/hip_bf16.h>

#define GD 256           // feature dim
#define TILE_ROWS 32     // rows per GEMM block
#define LDS_STRIDE 260   // 256 + 4 pad -> conflict-free column reads

typedef float v2f __attribute__((ext_vector_type(2)));
typedef float v8f __attribute__((ext_vector_type(8)));

// ---------------- small elementwise kernels ----------------

__global__ void gcn_zero_kernel(float* __restrict__ p, int n) {
    int i = blockIdx.x * blockDim.x + threadIdx.x;
    if (i < n) p[i] = 0.0f;
}

__global__ void gcn_deg_kernel(const int* __restrict__ dst, float* __restrict__ deg, int ne) {
    int e = blockIdx.x * blockDim.x + threadIdx.x;
    if (e < ne) unsafeAtomicAdd(&deg[dst[e]], 1.0f);
}

__global__ void gcn_dinv_kernel(const float* __restrict__ deg, float* __restrict__ dinv, int n) {
    int i = blockIdx.x * blockDim.x + threadIdx.x;
    if (i < n) dinv[i] = rsqrtf(deg[i] + 1.0f);   // +1 for self-loop; always > 0
}

__global__ void gcn_norm_kernel(const int* __restrict__ src, const int* __restrict__ dst,
                                const float* __restrict__ dinv, float* __restrict__ norm, int ne) {
    int e = blockIdx.x * blockDim.x + threadIdx.x;
    if (e < ne) norm[e] = dinv[src[e]] * dinv[dst[e]];
}

// Repack W[256x256] row-major into float2 pairs along K:
// Wp[(k2)*GD + n] = { W[2*k2][n], W[2*k2+1][n] }  => B fragment = one global_load_b64
__global__ void gcn_repack_kernel(const float* __restrict__ W, float2* __restrict__ Wp) {
    int gid = blockIdx.x * blockDim.x + threadIdx.x;      // [0, GD/2 * GD)
    if (gid >= (GD / 2) * GD) return;
    int k2 = gid >> 8;            // 0..127
    int n  = gid & (GD - 1);
    int kk = k2 * 2;
    Wp[gid] = make_float2(W[(size_t)kk * GD + n], W[(size_t)(kk + 1) * GD + n]);
}

// agg = h * dinv^2 + bias   (self-loop message + bias, dense init; one float4/thread)
__global__ void gcn_combine_kernel(const float* __restrict__ h, const float* __restrict__ dinv,
                                   const float* __restrict__ bias, float* __restrict__ agg, int nrows) {
    int gid = blockIdx.x * blockDim.x + threadIdx.x;
    int total = nrows * (GD / 4);
    if (gid >= total) return;
    int node = gid >> 6;            // GD/4 == 64 chunks per row
    int c = (gid & 63) * 4;
    float s = dinv[node]; s = s * s;
    float4 hv = *(const float4*)(h + (size_t)node * GD + c);
    float4 bv = *(const float4*)(bias + c);
    float4 o;
    o.x = hv.x * s + bv.x;
    o.y = hv.y * s + bv.y;
    o.z = hv.z * s + bv.z;
    o.w = hv.w * s + bv.w;
    *(float4*)(agg + (size_t)node * GD + c) = o;
}

// one wave per edge: agg[dst] += h[src] * norm  (8 floats per lane, fp32 HW atomics)
__global__ __launch_bounds__(256)
void gcn_scatter_kernel(const float* __restrict__ h, const float* __restrict__ norm,
                        const int* __restrict__ src, const int* __restrict__ dst,
                        float* __restrict__ agg, int ne) {
    int e = blockIdx.x * 8 + (threadIdx.x >> 5);
    if (e >= ne) return;
    int lane = threadIdx.x & 31;
    int s = src[e], d = dst[e];
    float w = norm[e];
    const float* hp = h + (size_t)s * GD + lane * 8;
    float* ap = agg + (size_t)d * GD + lane * 8;
    float4 v0 = *(const float4*)(hp);
    float4 v1 = *(const float4*)(hp + 4);
    unsafeAtomicAdd(ap + 0, v0.x * w);
    unsafeAtomicAdd(ap + 1, v0.y * w);
    unsafeAtomicAdd(ap + 2, v0.z * w);
    unsafeAtomicAdd(ap + 3, v0.w * w);
    unsafeAtomicAdd(ap + 4, v1.x * w);
    unsafeAtomicAdd(ap + 5, v1.y * w);
    unsafeAtomicAdd(ap + 6, v1.z * w);
    unsafeAtomicAdd(ap + 7, v1.w * w);
}

__global__ void gcn_relu_kernel(const float* __restrict__ in, float* __restrict__ out, int n4) {
    int i = blockIdx.x * blockDim.x + threadIdx.x;
    if (i >= n4) return;
    float4 v = *(const float4*)(in + (size_t)i * 4);
    v.x = fmaxf(v.x, 0.0f);
    v.y = fmaxf(v.y, 0.0f);
    v.z = fmaxf(v.z, 0.0f);
    v.w = fmaxf(v.w, 0.0f);
    *(float4*)(out + (size_t)i * 4) = v;
}

// ---------------- fp32 WMMA GEMM: H[nrows x 256] = X[nrows x 256] @ W[256 x 256] ----------------
// Block = 256 threads (8 waves). Block covers 32 rows; each wave owns a 32-wide column
// strip (two 16x16 tiles) x two 16-row tiles => 4 accumulators, K=256 in steps of 4.
// B comes from the K-paired repacked Wp (one b64 load per fragment).
__global__ __launch_bounds__(256)
void gcn_gemm_wmma_kernel(const float* __restrict__ X, const float2* __restrict__ Wp,
                          float* __restrict__ H, int nrows) {
    __shared__ float Xs[TILE_ROWS * LDS_STRIDE];
    const int tid = threadIdx.x;
    const int rowBase = blockIdx.x * TILE_ROWS;

    // cooperative load of the 32x256 X slab (8 float4 per thread)
    {
        const int r = tid >> 3;             // 0..31
        const int cBase = (tid & 7) * 32;   // 0,32,...,224
        const int gr = rowBase + r;
        float4* lp = (float4*)(Xs + r * LDS_STRIDE + cBase);
        if (gr < nrows) {
            const float4* gp = (const float4*)(X + (size_t)gr * GD + cBase);
#pragma unroll
            for (int j = 0; j < 8; ++j) lp[j] = gp[j];
        } else {
            float4 z = make_float4(0.f, 0.f, 0.f, 0.f);
#pragma unroll
            for (int j = 0; j < 8; ++j) lp[j] = z;
        }
    }
    __syncthreads();

    const int wave = tid >> 5;              // 0..7
    const int lane = tid & 31;
    const int m = lane & 15;                // row within tile (A) / col within tile (B)
    const int half = lane >> 4;             // upper half-wave holds K+2,K+3
    const int koff = half << 1;
    const int n0 = wave * 32;               // this wave's column strip

    v8f c00 = {}; v8f c01 = {}; v8f c10 = {}; v8f c11 = {};

    const float* xs0 = Xs + m * LDS_STRIDE + koff;          // row tile 0 (rows 0..15)
    const float* xs1 = Xs + (m + 16) * LDS_STRIDE + koff;   // row tile 1 (rows 16..31)
    // Wp rows come in (K/2) pairs; lane's pair row for k-step k4 is (2*k4 + half)
    const v2f* wp = (const v2f*)Wp + (size_t)half * GD + (n0 + m);

    for (int k4 = 0; k4 < GD / 4; ++k4) {
        v2f a0 = *(const v2f*)(xs0 + 4 * k4);   // ds_load_b64, conflict-free via pad
        v2f a1 = *(const v2f*)(xs1 + 4 * k4);
        v2f b0 = wp[(size_t)k4 * 2 * GD];        // global_load_b64
        v2f b1 = wp[(size_t)k4 * 2 * GD + 16];   // same clause, +128B immediate
        c00 = __builtin_amdgcn_wmma_f32_16x16x4_f32(false, a0, false, b0, (short)0, c00, false, false);
        c01 = __builtin_amdgcn_wmma_f32_16x16x4_f32(false, a0, false, b1, (short)0, c01, false, false);
        c10 = __builtin_amdgcn_wmma_f32_16x16x4_f32(false, a1, false, b0, (short)0, c10, false, false);
        c11 = __builtin_amdgcn_wmma_f32_16x16x4_f32(false, a1, false, b1, (short)0, c11, false, false);
    }

    // store: lanes 0-15 hold M = j, lanes 16-31 hold M = j+8; N = n0 + (lane&15)
    const int colA = n0 + m;
    const int rOff = half * 8;
    if (rowBase + TILE_ROWS <= nrows) {
        // fast path: one base pointer, constant immediate offsets -> store clause
        float* p0 = H + (size_t)(rowBase + rOff) * GD + colA;
#pragma unroll
        for (int j = 0; j < 8; ++j) {
            p0[(size_t)j * GD]             = c00[j];
            p0[(size_t)j * GD + 16]        = c01[j];
            p0[(size_t)(j + 16) * GD]      = c10[j];
            p0[(size_t)(j + 16) * GD + 16] = c11[j];
        }
    } else {
#pragma unroll
        for (int j = 0; j < 8; ++j) {
            int r0 = rowBase + rOff + j;
            int r1 = rowBase + 16 + rOff + j;
            if (r0 < nrows) {
                H[(size_t)r0 * GD + colA]      = c00[j];
                H[(size_t)r0 * GD + colA + 16] = c01[j];
            }
            if (r1 < nrows) {
                H[(size_t)r1 * GD + colA]      = c10[j];
                H[(size_t)r1 * GD + colA + 16] = c11[j];
            }
        }
    }
}

// ---------------- driver ----------------

extern "C" void kernel_launch(void* const* d_in, const int* in_sizes, int n_in,
                              void* d_out, int out_size, void* d_ws, size_t ws_size,
                              hipStream_t stream) {
    const float* x   = (const float*)d_in[0];
    const int*   ei  = (const int*)d_in[1];
    const float* W1  = (const float*)d_in[2];
    const float* b1  = (const float*)d_in[3];
    const float* W2  = (const float*)d_in[4];
    const float* b2  = (const float*)d_in[5];

    const int N  = in_sizes[0] / GD;
    const int E  = in_sizes[1] / 2;
    const int* src = ei;          // edge_index[0,:]
    const int* dst = ei + E;      // edge_index[1,:]

    float* ws   = (float*)d_ws;
    float* h    = ws;                          // N*GD
    float* agg  = ws + (size_t)N * GD;         // N*GD
    float* dinv = ws + (size_t)2 * N * GD;     // N
    float* norm = dinv + N;                    // E
    float* deg  = norm + E;                    // N
    float2* Wp  = (float2*)(deg + N);          // (GD/2)*GD float2 = 256KB
    float* out  = (float*)d_out;               // N*GD, also layer-1 activation

    const int T = 256;
    const int nBlk = (N + T - 1) / T;
    const int eBlk = (E + T - 1) / T;
    const int gBlk = (N + TILE_ROWS - 1) / TILE_ROWS;
    const int cBlk = (N * (GD / 4) + T - 1) / T;
    const int sBlk = (E + 7) / 8;
    const int pBlk = ((GD / 2) * GD + T - 1) / T;

    // normalization: deg -> dinv -> per-edge norm
    gcn_zero_kernel<<<nBlk, T, 0, stream>>>(deg, N);
    gcn_deg_kernel<<<eBlk, T, 0, stream>>>(dst, deg, E);
    gcn_dinv_kernel<<<nBlk, T, 0, stream>>>(deg, dinv, N);
    gcn_norm_kernel<<<eBlk, T, 0, stream>>>(src, dst, dinv, norm, E);

    // ---- layer 1 ----
    gcn_repack_kernel<<<pBlk, T, 0, stream>>>(W1, Wp);
    gcn_gemm_wmma_kernel<<<gBlk, T, 0, stream>>>(x, Wp, h, N);
    gcn_combine_kernel<<<cBlk, T, 0, stream>>>(h, dinv, b1, agg, N);
    gcn_scatter_kernel<<<sBlk, T, 0, stream>>>(h, norm, src, dst, agg, E);
    gcn_relu_kernel<<<cBlk, T, 0, stream>>>(agg, out, N * (GD / 4));

    // ---- layer 2 ----
    gcn_repack_kernel<<<pBlk, T, 0, stream>>>(W2, Wp);
    gcn_gemm_wmma_kernel<<<gBlk, T, 0, stream>>>(out, Wp, h, N);
    gcn_combine_kernel<<<cBlk, T, 0, stream>>>(h, dinv, b2, agg, N);
    gcn_scatter_kernel<<<sBlk, T, 0, stream>>>(h, norm, src, dst, agg, E);
    gcn_relu_kernel<<<cBlk, T, 0, stream>>>(agg, out, N * (GD / 4));
}